// SimpleTGAT_26903675142751
// MI455X (gfx1250) — compile-verified
//
#include <hip/hip_runtime.h>
#include <hip/hip_bf16.h>

typedef __attribute__((ext_vector_type(2))) float v2f;
typedef __attribute__((ext_vector_type(4))) float v4f;
typedef __attribute__((ext_vector_type(8))) float v8f;

#define DD 128
#define HH 4
#define CC 32
#define ED 16

// ---------- helpers ----------
__device__ __forceinline__ unsigned int fenc(float f) {
    unsigned int b = __float_as_uint(f);
    return (b & 0x80000000u) ? ~b : (b | 0x80000000u);
}
__device__ __forceinline__ float fdec(unsigned int u) {
    unsigned int b = (u & 0x80000000u) ? (u & 0x7FFFFFFFu) : ~u;
    return __uint_as_float(b);
}

__global__ void k_fill(float* __restrict__ p, float v, long n) {
    long i = (long)blockIdx.x * blockDim.x + threadIdx.x;
    if (i < n) p[i] = v;
}

// ---------- degree + mean incoming edge_attr (self-loop fill value) ----------
__global__ void k_deg_attr(const int* __restrict__ dst, const float* __restrict__ eattr,
                           float* __restrict__ deg, float* __restrict__ loopA, int E) {
    int e = blockIdx.x * blockDim.x + threadIdx.x;
    if (e >= E) return;
    int d = dst[e];
    atomicAdd(&deg[d], 1.0f);
    const float* ea = eattr + (size_t)e * ED;
    float* la = loopA + (size_t)d * ED;
#pragma unroll
    for (int j = 0; j < ED; ++j) atomicAdd(&la[j], ea[j]);
}

__global__ void k_div_loop(const float* __restrict__ deg, float* __restrict__ loopA, int N) {
    int n = blockIdx.x * blockDim.x + threadIdx.x;
    if (n >= N) return;
    float s = fmaxf(deg[n], 1.0f);
#pragma unroll
    for (int j = 0; j < ED; ++j) loopA[(size_t)n * ED + j] /= s;
}

// ---------- WMMA f32 GEMM:  C[M x 128] = A[M x K] @ W[K x 128] + bias ----------
// One wave computes a 32-row x 64-col strip: two 16-row tiles share B frags,
// 8 v8f accumulators (64 VGPRs) so nothing spills. Two waves per row strip
// cover the 128 columns. Requires M % 16 == 0 (true for all call sites); the
// row tail is gated wave-uniformly so EXEC is all-ones through every WMMA.
__global__ void k_gemm(const float* __restrict__ A, const float* __restrict__ W,
                       const float* __restrict__ bias, float* __restrict__ C,
                       int M, int K) {
    int wave = (blockIdx.x * blockDim.x + threadIdx.x) >> 5;
    int lane = threadIdx.x & 31;
    int nstrips = (M + 31) >> 5;
    if (wave >= nstrips * 2) return;       // wave-uniform
    int colh = (wave & 1) * 64;            // column half: 0 or 64
    long row0 = (long)(wave >> 1) * 32;
    bool two = (row0 + 16) < M;            // wave-uniform second-row-tile gate
    int ml = lane & 15;                    // M index for A, N index for B/C
    int kh = lane >> 4;                    // which K pair this half-wave holds
    long rowA0 = row0 + ml;
    long rowA1 = rowA0 + 16;

    v8f acc0[4], acc1[4];
#pragma unroll
    for (int t = 0; t < 4; ++t)
#pragma unroll
        for (int r = 0; r < 8; ++r) { acc0[t][r] = 0.0f; acc1[t][r] = 0.0f; }

    for (int k0 = 0; k0 < K; k0 += 4) {
        int ka = k0 + 2 * kh;
        v2f a0 = *(const v2f*)(A + rowA0 * K + ka);        // global_load_b64
        v2f a1;
        if (two) a1 = *(const v2f*)(A + rowA1 * K + ka);
        else { a1[0] = 0.0f; a1[1] = 0.0f; }
#pragma unroll
        for (int t = 0; t < 4; ++t) {
            int col = colh + t * 16 + ml;
            v2f b;
            b[0] = W[(size_t)ka * DD + col];
            b[1] = W[(size_t)(ka + 1) * DD + col];
            acc0[t] = __builtin_amdgcn_wmma_f32_16x16x4_f32(
                false, a0, false, b, (short)0, acc0[t], false, false);
            acc1[t] = __builtin_amdgcn_wmma_f32_16x16x4_f32(
                false, a1, false, b, (short)0, acc1[t], false, false);
        }
    }
#pragma unroll
    for (int t = 0; t < 4; ++t) {
        int col = colh + t * 16 + ml;
        float bc = bias[col];
#pragma unroll
        for (int r = 0; r < 8; ++r) {
            long rr = row0 + r + 8 * kh;   // C/D layout: hi half-wave holds M+8
            C[rr * DD + col] = acc0[t][r] + bc;
        }
    }
    if (two) {
#pragma unroll
        for (int t = 0; t < 4; ++t) {
            int col = colh + t * 16 + ml;
            float bc = bias[col];
#pragma unroll
            for (int r = 0; r < 8; ++r) {
                long rr = row0 + 16 + r + 8 * kh;
                C[rr * DD + col] = acc1[t][r] + bc;
            }
        }
    }
}

// ---------- edge logits (wave per edge; lane owns 4 consecutive channels) ----
// head = lane>>3 (8 lanes x 4 ch = 32 ch per head); xl/xr gathers are b128.
__global__ void k_edge_logit(const int* __restrict__ srcI, const int* __restrict__ dstI,
                             const float* __restrict__ eattr, const float* __restrict__ loopA,
                             const float* __restrict__ xl, const float* __restrict__ xr,
                             const float* __restrict__ We, const float* __restrict__ att,
                             float* __restrict__ logit, unsigned int* __restrict__ nmax,
                             int E, int NE) {
    int e = (blockIdx.x * blockDim.x + threadIdx.x) >> 5;
    int lane = threadIdx.x & 31;
    if (e >= NE) return;
    int s, d; const float* ea;
    if (e < E) { s = srcI[e]; d = dstI[e]; ea = eattr + (size_t)e * ED; }
    else       { s = e - E;   d = s;       ea = loopA + (size_t)s * ED; }

    int ch0 = lane * 4;
    v4f xlv = *(const v4f*)(xl + (size_t)s * DD + ch0);    // global_load_b128
    v4f xrv = *(const v4f*)(xr + (size_t)d * DD + ch0);
    v4f eev;
#pragma unroll
    for (int j = 0; j < 4; ++j) eev[j] = 0.0f;
#pragma unroll
    for (int j = 0; j < ED; ++j) {
        float aj = ea[j];
        v4f wv = *(const v4f*)(We + (size_t)j * DD + ch0);
#pragma unroll
        for (int c = 0; c < 4; ++c) eev[c] += aj * wv[c];
    }
    v4f attv = *(const v4f*)(att + ch0);
    float part = 0.0f;
#pragma unroll
    for (int c = 0; c < 4; ++c) {
        float m = xlv[c] + xrv[c] + eev[c];
        m = (m > 0.0f) ? m : 0.2f * m;                     // LeakyReLU(0.2)
        part += m * attv[c];
    }
    // reduce within each 8-lane head group
#pragma unroll
    for (int off = 1; off < 8; off <<= 1) part += __shfl_xor(part, off, 32);
    if ((lane & 7) == 0) {
        int h = lane >> 3;
        logit[(size_t)e * HH + h] = part;
        atomicMax(&nmax[(size_t)d * HH + h], fenc(part));
    }
}

// ---------- exp(logit - max) and denominator ----------
__global__ void k_exp_den(const int* __restrict__ dstI, float* __restrict__ logit,
                          const unsigned int* __restrict__ nmax, float* __restrict__ den,
                          int E, int NE) {
    int e = blockIdx.x * blockDim.x + threadIdx.x;
    if (e >= NE) return;
    int d = (e < E) ? dstI[e] : (e - E);
#pragma unroll
    for (int h = 0; h < HH; ++h) {
        float mx = fdec(nmax[(size_t)d * HH + h]);
        float ex = expf(logit[(size_t)e * HH + h] - mx);
        logit[(size_t)e * HH + h] = ex;              // reuse buffer for exp value
        atomicAdd(&den[(size_t)d * HH + h], ex);
    }
}

// ---------- normalized scatter accumulation: h[dst] += a * xl[src] ----------
__global__ void k_accum(const int* __restrict__ srcI, const int* __restrict__ dstI,
                        const float* __restrict__ logit, const float* __restrict__ den,
                        const float* __restrict__ xl, float* __restrict__ hb,
                        int E, int NE) {
    int e = (blockIdx.x * blockDim.x + threadIdx.x) >> 5;
    int lane = threadIdx.x & 31;
    if (e >= NE) return;
    int s, d;
    if (e < E) { s = srcI[e]; d = dstI[e]; } else { s = e - E; d = s; }
    int ch0 = lane * 4;
    int h = lane >> 3;
    float a = logit[(size_t)e * HH + h] / den[(size_t)d * HH + h];
    v4f xlv = *(const v4f*)(xl + (size_t)s * DD + ch0);    // global_load_b128
    float* hp = hb + (size_t)d * DD + ch0;
#pragma unroll
    for (int c = 0; c < 4; ++c) atomicAdd(&hp[c], a * xlv[c]);
}

// ---------- conv bias + ELU ----------
__global__ void k_elu(float* __restrict__ hb, const float* __restrict__ cb, long n128) {
    long i = (long)blockIdx.x * blockDim.x + threadIdx.x;
    if (i >= n128) return;
    float v = hb[i] + cb[i & (DD - 1)];
    hb[i] = (v > 0.0f) ? v : (expf(v) - 1.0f);
}

// ---------- LayerNorm + MLP head (wave per node, LDS staging) ----------
__global__ void k_final(const float* __restrict__ hb,
                        const float* __restrict__ ln_g, const float* __restrict__ ln_b,
                        const float* __restrict__ h1w, const float* __restrict__ h1b,
                        const float* __restrict__ h2w, const float* __restrict__ h2b,
                        float* __restrict__ out, int N) {
    __shared__ float sh[8][DD];
    __shared__ float sy[8][64];
    int wid = threadIdx.x >> 5;
    int lane = threadIdx.x & 31;
    int node = blockIdx.x * 8 + wid;
    if (node >= N) return;                    // wave-uniform exit, no barriers used

    float v[4];
#pragma unroll
    for (int r = 0; r < 4; ++r) v[r] = hb[(size_t)node * DD + r * 32 + lane];
    float s = v[0] + v[1] + v[2] + v[3];
#pragma unroll
    for (int off = 16; off > 0; off >>= 1) s += __shfl_xor(s, off, 32);
    float mu = s / 128.0f;
    float q = 0.0f;
#pragma unroll
    for (int r = 0; r < 4; ++r) { float dv = v[r] - mu; q += dv * dv; }
#pragma unroll
    for (int off = 16; off > 0; off >>= 1) q += __shfl_xor(q, off, 32);
    float inv = rsqrtf(q / 128.0f + 1e-5f);
#pragma unroll
    for (int r = 0; r < 4; ++r) {
        int ch = r * 32 + lane;
        sh[wid][ch] = (v[r] - mu) * inv * ln_g[ch] + ln_b[ch];
    }
    // intra-wave LDS RAW: DS ops are in-order within a wave (DScnt), no barrier needed
#pragma unroll
    for (int t = 0; t < 2; ++t) {
        int j = t * 32 + lane;
        float acc = h1b[j];
        for (int k = 0; k < DD; ++k) acc += sh[wid][k] * h1w[k * 64 + j];
        sy[wid][j] = fmaxf(acc, 0.0f);
    }
    if (lane < 3) {
        float acc = h2b[lane];
        for (int k = 0; k < 64; ++k) acc += sy[wid][k] * h2w[k * 3 + lane];
        out[(size_t)node * 3 + lane] = acc;
    }
}

static inline int cdiv(long a, long b) { return (int)((a + b - 1) / b); }

extern "C" void kernel_launch(void* const* d_in, const int* in_sizes, int n_in,
                              void* d_out, int out_size, void* d_ws, size_t ws_size,
                              hipStream_t stream) {
    const float* x     = (const float*)d_in[0];
    const int*   ei    = (const int*)d_in[1];
    const float* eattr = (const float*)d_in[2];
    const float* in_w  = (const float*)d_in[3];
    const float* in_b  = (const float*)d_in[4];
    const float* Wl    = (const float*)d_in[5];
    const float* bl    = (const float*)d_in[6];
    const float* Wr    = (const float*)d_in[7];
    const float* br    = (const float*)d_in[8];
    const float* We    = (const float*)d_in[9];
    const float* att   = (const float*)d_in[10];
    const float* cb    = (const float*)d_in[11];
    const float* ln_g  = (const float*)d_in[12];
    const float* ln_b  = (const float*)d_in[13];
    const float* h1w   = (const float*)d_in[14];
    const float* h1b   = (const float*)d_in[15];
    const float* h2w   = (const float*)d_in[16];
    const float* h2b   = (const float*)d_in[17];

    const int N  = in_sizes[0] / DD;     // 50000
    const int E  = in_sizes[2] / ED;     // 600000
    const int NE = E + N;                // with self loops
    const int* srcI = ei;
    const int* dstI = ei + E;

    // workspace layout (floats)
    float* ws    = (float*)d_ws;
    float* deg   = ws;                                    // N
    float* loopA = deg + N;                               // N*16
    float* hbuf  = loopA + (size_t)N * ED;                // N*128
    float* xl    = hbuf + (size_t)N * DD;                 // N*128
    float* xr    = xl + (size_t)N * DD;                   // N*128
    float* logit = xr + (size_t)N * DD;                   // NE*4
    unsigned int* nmax = (unsigned int*)(logit + (size_t)NE * HH); // N*4
    float* den   = (float*)(nmax + (size_t)N * HH);       // N*4

    const int B = 256;

    // self-loop attrs = mean of incoming edge_attr
    k_fill<<<cdiv((long)N * 17, B), B, 0, stream>>>(deg, 0.0f, (long)N * 17);
    k_deg_attr<<<cdiv(E, B), B, 0, stream>>>(dstI, eattr, deg, loopA, E);
    k_div_loop<<<cdiv(N, B), B, 0, stream>>>(deg, loopA, N);

    // input projection: h = x @ in_w + in_b
    int nstrips = cdiv(N, 32);
    int gemmBlocks = cdiv((long)nstrips * 2, 8);  // 2 column-half waves per strip
    k_gemm<<<gemmBlocks, B, 0, stream>>>(x, in_w, in_b, hbuf, N, DD);

    for (int l = 0; l < 2; ++l) {
        k_gemm<<<gemmBlocks, B, 0, stream>>>(hbuf, Wl + (size_t)l * DD * DD, bl + l * DD, xl, N, DD);
        k_gemm<<<gemmBlocks, B, 0, stream>>>(hbuf, Wr + (size_t)l * DD * DD, br + l * DD, xr, N, DD);

        k_fill<<<cdiv((long)N * 8, B), B, 0, stream>>>((float*)nmax, 0.0f, (long)N * 8); // nmax+den

        k_edge_logit<<<cdiv(NE, 8), B, 0, stream>>>(srcI, dstI, eattr, loopA, xl, xr,
                                                    We + (size_t)l * ED * DD, att + l * HH * CC,
                                                    logit, nmax, E, NE);
        k_exp_den<<<cdiv(NE, B), B, 0, stream>>>(dstI, logit, nmax, den, E, NE);

        k_fill<<<cdiv((long)N * DD, B), B, 0, stream>>>(hbuf, 0.0f, (long)N * DD);
        k_accum<<<cdiv(NE, 8), B, 0, stream>>>(srcI, dstI, logit, den, xl, hbuf, E, NE);
        k_elu<<<cdiv((long)N * DD, B), B, 0, stream>>>(hbuf, cb + l * DD, (long)N * DD);
    }

    k_final<<<cdiv(N, 8), B, 0, stream>>>(hbuf, ln_g, ln_b, h1w, h1b, h2w, h2b,
                                          (float*)d_out, N);
}